// GATNetwork_32985348833682
// MI455X (gfx1250) — compile-verified
//
#include <hip/hip_runtime.h>

#define NF 128      // feature / hidden width
#define KD 128      // GEMM K
#define KC 32       // K chunk staged in LDS
#define AST 36      // LDS row stride for A chunk (32 + 4 pad via TDM pad_enable)
// B LDS row stride = Ncols + 16 (TDM pad: 16 dwords after every Ncols dwords)

typedef __attribute__((ext_vector_type(2))) float v2f;
typedef __attribute__((ext_vector_type(8))) float v8f;
typedef __attribute__((ext_vector_type(4))) unsigned tdm_g0_t;
typedef __attribute__((ext_vector_type(8))) int tdm_g1_t;
typedef __attribute__((ext_vector_type(4))) int tdm_g2_t;

// ---------------------------------------------------------------------------
// TDM 2D tile load: Global -> LDS, fp32 elements, optional LDS row padding.
// Builds the D# per CDNA5 ISA ch.8 (groups 2/3 zero => 2D tensor).
// ---------------------------------------------------------------------------
__device__ __forceinline__ void tdm_load_2d(const void* gaddr, unsigned lds_byte_off,
                                            unsigned tensor_d0, unsigned tensor_d1,
                                            unsigned stride_d0, unsigned tile_d0,
                                            unsigned tile_d1, unsigned pad_int_code,
                                            unsigned pad_amt_code) {
  unsigned long long ga = (unsigned long long)(uintptr_t)gaddr;
  tdm_g0_t g0;
  g0[0] = 1u;                                              // count=1 (valid user D#)
  g0[1] = lds_byte_off;                                    // lds_addr
  g0[2] = (unsigned)(ga & 0xFFFFFFFFu);                    // global_addr[31:0]
  g0[3] = (unsigned)((ga >> 32) & 0x01FFFFFFu) | (2u << 30); // global_addr[56:32] | type=2
  tdm_g1_t g1;
  g1[0] = (int)((2u << 16)                                 // data_size = 4B
              | (1u << 20)                                 // pad_enable
              | (pad_int_code << 22)                       // pad interval (2^(c+1) dwords)
              | (pad_amt_code << 25));                     // pad amount (c+1 dwords)
  g1[1] = (int)((tensor_d0 & 0xFFFFu) << 16);              // abar_addr=0 | tensor_dim0[15:0]
  g1[2] = (int)(((tensor_d0 >> 16) & 0xFFFFu) | ((tensor_d1 & 0xFFFFu) << 16));
  g1[3] = (int)(((tensor_d1 >> 16) & 0xFFFFu) | ((tile_d0 & 0xFFFFu) << 16));
  g1[4] = (int)(tile_d1 & 0xFFFFu);                        // tile_dim1 | tile_dim2=0
  g1[5] = (int)stride_d0;                                  // tensor_dim0_stride[31:0]
  g1[6] = 0;                                               // stride hi | dim1_stride lo
  g1[7] = 0;
  tdm_g2_t gz = {0, 0, 0, 0};
#if __clang_major__ >= 23
  tdm_g1_t gz8 = {0, 0, 0, 0, 0, 0, 0, 0};
  __builtin_amdgcn_tensor_load_to_lds(g0, g1, gz, gz, gz8, 0);
#else
  __builtin_amdgcn_tensor_load_to_lds(g0, g1, gz, gz, 0);
#endif
}

__device__ __forceinline__ unsigned lds_off(const void* p) {
  return (unsigned)(uintptr_t)p;   // LDS aperture: addr[31:0] == workgroup LDS offset
}

// ---------------------------------------------------------------------------
// C[M,Ncols] = leaky(A[M,128] @ B[128,Ncols] + bias, slope)
// fp32 WMMA 16x16x4; TDM double-buffered LDS staging (wave0 issues, TENSORcnt).
// 8 waves/block: wave = one 16-col tile, 4 row tiles (64 rows/block).
// ---------------------------------------------------------------------------
__launch_bounds__(256)
__global__ void gemm_wmma_tdm(const float* __restrict__ A, const float* __restrict__ B,
                              float* __restrict__ C, int M, int Ncols, int lnc,
                              const float* __restrict__ bias, float slope) {
  __shared__ float Alds[2][64 * AST];          // 2 x 9216 B
  __shared__ float Blds[2][KC * (NF + 16)];    // 2 x 18432 B  (total ~54 KB)
  const int t    = threadIdx.x;
  const int wave = t >> 5;
  const int lane = t & 31;
  const int m    = lane & 15;   // row-in-tile (A) / col-in-tile (B,C)
  const int g    = lane >> 4;   // half-wave select
  const int row0 = blockIdx.x * 64;
  const int ntiles = Ncols >> 4;
  const int bst  = Ncols + 16;  // padded B row stride in LDS
  const int ct   = wave;

  v8f zero = {0.f, 0.f, 0.f, 0.f, 0.f, 0.f, 0.f, 0.f};
  v8f acc[4];
  acc[0] = zero; acc[1] = zero; acc[2] = zero; acc[3] = zero;

  const int coff = (ct < ntiles ? ct : 0) << 4;

  const unsigned aoff0 = lds_off(&Alds[0][0]), aoff1 = lds_off(&Alds[1][0]);
  const unsigned boff0 = lds_off(&Blds[0][0]), boff1 = lds_off(&Blds[1][0]);
  const unsigned rows_avail = (unsigned)(M - row0);   // grid guarantees > 0
  const unsigned un = (unsigned)Ncols;

  // prologue: wave0 kicks TDM for chunk 0 into buffer 0
  if (t < 32) {
    tdm_load_2d(A + (size_t)row0 * KD, aoff0, 32u, rows_avail, (unsigned)KD,
                32u, 64u, 4u, 3u);                       // pad 4 dw every 32 dw -> AST=36
    tdm_load_2d(B, boff0, un, 32u, un, un, 32u,
                (unsigned)(lnc - 1), 15u);               // pad 16 dw every Ncols dw
  }

  const int nch = KD / KC;  // 4 chunks
  for (int i = 0; i < nch; ++i) {
    const int buf = i & 1;
    if (t < 32) {
      if (i + 1 < nch) {       // prefetch next chunk into the other buffer
        const int kc = (i + 1) * KC;
        tdm_load_2d(A + (size_t)row0 * KD + kc, (buf ? aoff0 : aoff1),
                    32u, rows_avail, (unsigned)KD, 32u, 64u, 4u, 3u);
        tdm_load_2d(B + (size_t)kc * Ncols, (buf ? boff0 : boff1),
                    un, 32u, un, un, 32u, (unsigned)(lnc - 1), 15u);
        __builtin_amdgcn_s_wait_tensorcnt(2);  // in-order TDM: chunk i resident
      } else {
        __builtin_amdgcn_s_wait_tensorcnt(0);
      }
    }
    __syncthreads();           // chunk i visible to all waves

    const float* __restrict__ Ab = Alds[buf];
    const float* __restrict__ Bb = Blds[buf];
    #pragma unroll
    for (int k0 = 0; k0 < KC; k0 += 4) {
      const int ka = k0 + 2 * g;
      v2f b;
      b.x = Bb[ka * bst + coff + m];
      b.y = Bb[(ka + 1) * bst + coff + m];
      #pragma unroll
      for (int rt = 0; rt < 4; ++rt) {
        v2f a;
        a.x = Ab[(rt * 16 + m) * AST + ka];
        a.y = Ab[(rt * 16 + m) * AST + ka + 1];
        acc[rt] = __builtin_amdgcn_wmma_f32_16x16x4_f32(
            false, a, false, b, (short)0, acc[rt], false, false);
      }
    }
    __syncthreads();           // all waves done reading buf before TDM reuses it
  }

  if (ct < ntiles) {
    const int col = (ct << 4) + m;
    const float bv = bias ? bias[col] : 0.0f;
    #pragma unroll
    for (int rt = 0; rt < 4; ++rt) {
      #pragma unroll
      for (int r = 0; r < 8; ++r) {
        int row = row0 + rt * 16 + r + 8 * g;   // D VGPR r <-> matrix row r + 8*g
        if (row < M) {
          float v = acc[rt][r] + bv;
          v = v > 0.0f ? v : slope * v;
          C[(size_t)row * Ncols + col] = v;
        }
      }
    }
  }
}

// ---------------------------------------------------------------------------
// Per-node attention scores: a_s[i] = hW[i].att_src, a_d[i] = hW[i].att_dst
// ---------------------------------------------------------------------------
__global__ void score_kernel(const float* __restrict__ hW, const float* __restrict__ att_s,
                             const float* __restrict__ att_d, float* __restrict__ a_s,
                             float* __restrict__ a_d, int n) {
  int node = (blockIdx.x * blockDim.x + threadIdx.x) >> 5;
  int lane = threadIdx.x & 31;
  if (node >= n) return;
  float4 v  = ((const float4*)(hW + (size_t)node * NF))[lane];
  float4 s4 = ((const float4*)att_s)[lane];
  float4 d4 = ((const float4*)att_d)[lane];
  float ss = v.x * s4.x + v.y * s4.y + v.z * s4.z + v.w * s4.w;
  float sd = v.x * d4.x + v.y * d4.y + v.z * d4.z + v.w * d4.w;
  for (int off = 16; off; off >>= 1) {
    ss += __shfl_xor(ss, off, 32);
    sd += __shfl_xor(sd, off, 32);
  }
  if (lane == 0) { a_s[node] = ss; a_d[node] = sd; }
}

// monotone float<->uint map so segment-max can use atomicMax(u32)
__device__ __forceinline__ unsigned f2mono(float x) {
  unsigned b = __float_as_uint(x);
  return (b & 0x80000000u) ? ~b : (b | 0x80000000u);
}
__device__ __forceinline__ float mono2f(unsigned u) {
  unsigned b = (u & 0x80000000u) ? (u & 0x7FFFFFFFu) : ~u;
  return __uint_as_float(b);
}

__global__ void layer_init(float* __restrict__ hacc, const float* __restrict__ bias,
                           unsigned* __restrict__ mtrans, float* __restrict__ ssum, int n) {
  int idx = blockIdx.x * blockDim.x + threadIdx.x;
  if (idx < n * NF) hacc[idx] = bias[idx & (NF - 1)];
  if (idx < n) { mtrans[idx] = 0x007FFFFFu; ssum[idx] = 0.0f; }  // map(-inf)
}

__global__ void edge_logits(const int* __restrict__ src, const int* __restrict__ dst,
                            const float* __restrict__ a_s, const float* __restrict__ a_d,
                            float* __restrict__ ebuf, unsigned* __restrict__ mtrans, int E_) {
  int e = blockIdx.x * blockDim.x + threadIdx.x;
  if (e >= E_) return;
  float v = a_s[src[e]] + a_d[dst[e]];
  v = v > 0.0f ? v : 0.2f * v;          // leaky_relu 0.2
  ebuf[e] = v;
  atomicMax(&mtrans[dst[e]], f2mono(v));
}

__global__ void edge_exp(const int* __restrict__ dst, float* __restrict__ ebuf,
                         const unsigned* __restrict__ mtrans, float* __restrict__ ssum, int E_) {
  int e = blockIdx.x * blockDim.x + threadIdx.x;
  if (e >= E_) return;
  int d = dst[e];
  float ex = __expf(ebuf[e] - mono2f(mtrans[d]));
  ebuf[e] = ex;
  atomicAdd(&ssum[d], ex);
}

// wave32 per edge: hacc[dst] += hW[src] * alpha  (gathers hit L2: hW = 51MB < 192MB)
__global__ void edge_scatter(const int* __restrict__ src, const int* __restrict__ dst,
                             const float* __restrict__ ebuf, const float* __restrict__ ssum,
                             const float* __restrict__ hW, float* __restrict__ hacc, int E_) {
  int e = (blockIdx.x * blockDim.x + threadIdx.x) >> 5;
  int lane = threadIdx.x & 31;
  if (e >= E_) return;
  int sN = src[e], dN = dst[e];
  float alpha = ebuf[e] / (ssum[dN] + 1e-16f);
  float4 v = ((const float4*)(hW + (size_t)sN * NF))[lane];
  float* out = hacc + (size_t)dN * NF + lane * 4;
  atomicAdd(out + 0, v.x * alpha);
  atomicAdd(out + 1, v.y * alpha);
  atomicAdd(out + 2, v.z * alpha);
  atomicAdd(out + 3, v.w * alpha);
}

// ------------------------- GraphNorm (per-feature over all N) ---------------
__global__ void gn_zero(float* cs, float* cs2) {
  int t = threadIdx.x;
  if (t < NF) { cs[t] = 0.0f; cs2[t] = 0.0f; }
}

__global__ void gn_colsum(const float* __restrict__ h, float* __restrict__ cs, int n) {
  int c = threadIdx.x & (NF - 1);
  int rows_per_blk = blockDim.x >> 7;
  int r0 = blockIdx.x * rows_per_blk + (threadIdx.x >> 7);
  int rs = gridDim.x * rows_per_blk;
  float acc = 0.0f;
  for (int r = r0; r < n; r += rs) {
    __builtin_prefetch(&h[(size_t)(r + rs) * NF + c], 0, 1);
    acc += h[(size_t)r * NF + c];
  }
  atomicAdd(&cs[c], acc);
}

__global__ void gn_mean(const float* cs, float* mean, int n) {
  int c = threadIdx.x;
  if (c < NF) mean[c] = cs[c] / (float)n;
}

__global__ void gn_colvar(const float* __restrict__ h, const float* __restrict__ mean,
                          const float* __restrict__ ms, float* __restrict__ cs2, int n) {
  int c = threadIdx.x & (NF - 1);
  int rows_per_blk = blockDim.x >> 7;
  int r0 = blockIdx.x * rows_per_blk + (threadIdx.x >> 7);
  int rs = gridDim.x * rows_per_blk;
  float mm = mean[c] * ms[c];
  float acc = 0.0f;
  for (int r = r0; r < n; r += rs) {
    float o = h[(size_t)r * NF + c] - mm;
    acc += o * o;
  }
  atomicAdd(&cs2[c], acc);
}

__global__ void gn_invstd(const float* cs2, float* inv, int n) {
  int c = threadIdx.x;
  if (c < NF) inv[c] = rsqrtf(cs2[c] / (float)n + 1e-5f);
}

__global__ void gn_apply(const float* __restrict__ h, const float* __restrict__ mean,
                         const float* __restrict__ ms, const float* __restrict__ inv,
                         const float* __restrict__ w, const float* __restrict__ b,
                         float* __restrict__ outp, int n) {
  int idx = blockIdx.x * blockDim.x + threadIdx.x;
  if (idx >= n * NF) return;
  int c = idx & (NF - 1);
  float o = h[idx] - mean[c] * ms[c];
  float v = w[c] * o * inv[c] + b[c];
  outp[idx] = v > 0.0f ? v : 0.01f * v;   // inter-layer leaky 0.01
}

// ------------------------------ pooling -------------------------------------
__global__ void pool_zero(float* pooled, int total) {
  int t = blockIdx.x * blockDim.x + threadIdx.x;
  if (t < total) pooled[t] = 0.0f;
}

__global__ void pool_sum(const float* __restrict__ h, const int* __restrict__ batch,
                         float* __restrict__ pooled, int n) {
  int idx = blockIdx.x * blockDim.x + threadIdx.x;
  if (idx >= n * NF) return;
  int r = idx >> 7, c = idx & (NF - 1);
  atomicAdd(&pooled[batch[r] * NF + c], h[idx]);
}

// ---------------------------------------------------------------------------
extern "C" void kernel_launch(void* const* d_in, const int* in_sizes, int n_in,
                              void* d_out, int out_size, void* d_ws, size_t ws_size,
                              hipStream_t stream) {
  const float* x       = (const float*)d_in[0];
  const int*   eidx    = (const int*)d_in[1];
  const int*   batch   = (const int*)d_in[2];
  const float* Ws      = (const float*)d_in[3];
  const float* att_src = (const float*)d_in[4];
  const float* att_dst = (const float*)d_in[5];
  const float* conv_b  = (const float*)d_in[6];
  const float* gn_w    = (const float*)d_in[7];
  const float* gn_b    = (const float*)d_in[8];
  const float* gn_ms   = (const float*)d_in[9];
  const float* W1      = (const float*)d_in[10];
  const float* b1      = (const float*)d_in[11];
  const float* W2      = (const float*)d_in[12];
  const float* b2      = (const float*)d_in[13];

  const int N = in_sizes[0] / NF;
  const int E = in_sizes[1] / 2;
  const int G = 64, Aact = 64, L = 3;

  float* ws = (float*)d_ws;
  size_t off = 0;
  float* hW     = ws + off; off += (size_t)N * NF;
  float* hacc   = ws + off; off += (size_t)N * NF;
  float* hnorm  = ws + off; off += (size_t)N * NF;
  float* ebuf   = ws + off; off += (size_t)E;
  float* a_s    = ws + off; off += N;
  float* a_d    = ws + off; off += N;
  float* ssum   = ws + off; off += N;
  unsigned* mtrans = (unsigned*)(ws + off); off += N;
  float* cs     = ws + off; off += NF;
  float* cs2    = ws + off; off += NF;
  float* mean   = ws + off; off += NF;
  float* inv    = ws + off; off += NF;
  float* pooled = ws + off; off += (size_t)G * NF;
  float* z      = ws + off; off += (size_t)G * NF;

  const int* src  = eidx;
  const int* dstv = eidx + E;

  const float* hin = x;
  for (int l = 0; l < L; ++l) {
    gemm_wmma_tdm<<<(N + 63) / 64, 256, 0, stream>>>(
        hin, Ws + (size_t)l * KD * NF, hW, N, NF, 7, nullptr, 1.0f);
    score_kernel<<<(N + 7) / 8, 256, 0, stream>>>(
        hW, att_src + l * NF, att_dst + l * NF, a_s, a_d, N);
    layer_init<<<(N * NF + 255) / 256, 256, 0, stream>>>(hacc, conv_b + l * NF, mtrans, ssum, N);
    edge_logits<<<(E + 255) / 256, 256, 0, stream>>>(src, dstv, a_s, a_d, ebuf, mtrans, E);
    edge_exp<<<(E + 255) / 256, 256, 0, stream>>>(dstv, ebuf, mtrans, ssum, E);
    edge_scatter<<<(E + 7) / 8, 256, 0, stream>>>(src, dstv, ebuf, ssum, hW, hacc, E);
    if (l < L - 1) {
      gn_zero<<<1, 128, 0, stream>>>(cs, cs2);
      gn_colsum<<<512, 256, 0, stream>>>(hacc, cs, N);
      gn_mean<<<1, 128, 0, stream>>>(cs, mean, N);
      gn_colvar<<<512, 256, 0, stream>>>(hacc, mean, gn_ms + l * NF, cs2, N);
      gn_invstd<<<1, 128, 0, stream>>>(cs2, inv, N);
      gn_apply<<<(N * NF + 255) / 256, 256, 0, stream>>>(
          hacc, mean, gn_ms + l * NF, inv, gn_w + l * NF, gn_b + l * NF, hnorm, N);
      hin = hnorm;
    }
  }

  pool_zero<<<(G * NF + 255) / 256, 256, 0, stream>>>(pooled, G * NF);
  pool_sum<<<(N * NF + 255) / 256, 256, 0, stream>>>(hacc, batch, pooled, N);
  // MLP head, also on the WMMA+TDM path
  gemm_wmma_tdm<<<1, 256, 0, stream>>>(pooled, W1, z, G, NF, 7, b1, 0.01f);
  gemm_wmma_tdm<<<1, 256, 0, stream>>>(z, W2, (float*)d_out, G, Aact, 6, b2, 1.0f);
}